// RNA_feature_extraction_77713138253983
// MI455X (gfx1250) — compile-verified
//
#include <hip/hip_runtime.h>

#define HIDDEN   64
#define NGRAPHS  64
#define LDSTRIDE 68   // 16-row tile row stride in LDS (padded vs 64 to dodge bank conflicts)

typedef __attribute__((ext_vector_type(2))) float v2f;
typedef __attribute__((ext_vector_type(8))) float v8f;

__device__ __forceinline__ void atomAddF(float* p, float v) {
    // hardware global_atomic_add_f32 (no CAS loop)
    unsafeAtomicAdd(p, v);
}

// h[n][j] = x[n] * ne_w[j] + ne_b[j]
__global__ __launch_bounds__(256) void node_encode(
    const float* __restrict__ x, const float* __restrict__ ne_w,
    const float* __restrict__ ne_b, float* __restrict__ h, int N)
{
    int i = blockIdx.x * blockDim.x + threadIdx.x;
    if (i < N * HIDDEN) {
        int n = i >> 6, j = i & 63;
        h[i] = x[n] * ne_w[j] + ne_b[j];
    }
}

// per edge k: m = relu(h[src] + edge_attr[k]*ee_w + ee_b); agg[dst] += m
// one wave32 per edge; lane handles channels (2*lane, 2*lane+1)
__global__ __launch_bounds__(256) void edge_scatter(
    const float* __restrict__ h, const int* __restrict__ ei,
    const float* __restrict__ ea, const float* __restrict__ ee_w,
    const float* __restrict__ ee_b, float* __restrict__ agg, int E)
{
    const int lane   = threadIdx.x & 31;
    const int wave   = (blockIdx.x * blockDim.x + threadIdx.x) >> 5;
    const int nWaves = (gridDim.x * blockDim.x) >> 5;
    const int c = lane * 2;
    const float2 w = *(const float2*)(ee_w + c);
    const float2 b = *(const float2*)(ee_b + c);
    for (int k = wave; k < E; k += nWaves) {
        const int   s = ei[k];
        const int   d = ei[E + k];
        const float a = ea[k];
        const float2 hv = *(const float2*)(h + (size_t)s * HIDDEN + c);
        float mx = hv.x + a * w.x + b.x; mx = mx > 0.f ? mx : 0.f;
        float my = hv.y + a * w.y + b.y; my = my > 0.f ? my : 0.f;
        float* p = agg + (size_t)d * HIDDEN + c;
        atomAddF(p,     mx);
        atomAddF(p + 1, my);
    }
}

// out = relu( relu((h+agg) @ w1 + b1) @ w2 + b2 )   — fused GINE node MLP + outer relu
// One wave per 16-row strip. K stepped by 4 via V_WMMA_F32_16X16X4_F32 (exact f32).
// A layout (16x4 f32): lanes 0-15 -> M=lane, K=k0,k0+1 ; lanes 16-31 -> M=lane-16, K=k0+2,k0+3.
// B layout (4x16):     lanes 0-15 -> N=lane, K=k0,k0+1 ; lanes 16-31 -> N=lane-16, K=k0+2,k0+3.
// C/D layout (16x16):  VGPR v -> M = v + 8*(lane>=16), N = lane&15.
__global__ __launch_bounds__(256) void gine_mlp_wmma(
    const float* __restrict__ h,  const float* __restrict__ agg,
    const float* __restrict__ w1, const float* __restrict__ b1,
    const float* __restrict__ w2, const float* __restrict__ b2,
    float* __restrict__ out, int nStrips)
{
    __shared__ float lds[8 * 16 * LDSTRIDE];   // 34.8 KB: per-wave 16x64 intermediate tile
    const int lane  = threadIdx.x & 31;
    const int wv    = threadIdx.x >> 5;
    const int strip = blockIdx.x * 8 + wv;
    if (strip >= nStrips) return;              // wave-uniform: EXEC stays all-ones for WMMA

    float* tbuf = lds + wv * (16 * LDSTRIDE);
    const int half = lane >> 4;                // 0 or 1
    const int l16  = lane & 15;
    const int koff = half * 2;
    const size_t rbase = ((size_t)strip * 16 + l16) * HIDDEN;

    // ---- GEMM 1: z = h + agg;  acc = z @ w1 ----
    v8f acc[4] = {};
    #pragma unroll
    for (int k0 = 0; k0 < HIDDEN; k0 += 4) {
        const int kb = k0 + koff;
        v2f a;
        a.x = h[rbase + kb]     + agg[rbase + kb];
        a.y = h[rbase + kb + 1] + agg[rbase + kb + 1];
        const float* w1r0 = w1 + (size_t)kb * HIDDEN;
        const float* w1r1 = w1r0 + HIDDEN;
        #pragma unroll
        for (int nt = 0; nt < 4; ++nt) {
            v2f b;
            b.x = w1r0[nt * 16 + l16];
            b.y = w1r1[nt * 16 + l16];
            acc[nt] = __builtin_amdgcn_wmma_f32_16x16x4_f32(
                false, a, false, b, (short)0, acc[nt], false, false);
        }
    }

    // bias + relu, C/D layout -> LDS (row-major 16x64 tile, padded stride)
    #pragma unroll
    for (int nt = 0; nt < 4; ++nt) {
        const float bias = b1[nt * 16 + l16];
        #pragma unroll
        for (int v = 0; v < 8; ++v) {
            float f = acc[nt][v] + bias;
            tbuf[(v + half * 8) * LDSTRIDE + nt * 16 + l16] = f > 0.f ? f : 0.f;
        }
    }

    // ---- GEMM 2: t @ w2 (A re-read from LDS in A layout) ----
    v8f dacc[4] = {};
    const float* trow = tbuf + l16 * LDSTRIDE;
    #pragma unroll
    for (int k0 = 0; k0 < HIDDEN; k0 += 4) {
        const int kb = k0 + koff;
        v2f a;
        a.x = trow[kb];
        a.y = trow[kb + 1];
        const float* w2r0 = w2 + (size_t)kb * HIDDEN;
        const float* w2r1 = w2r0 + HIDDEN;
        #pragma unroll
        for (int nt = 0; nt < 4; ++nt) {
            v2f b;
            b.x = w2r0[nt * 16 + l16];
            b.y = w2r1[nt * 16 + l16];
            dacc[nt] = __builtin_amdgcn_wmma_f32_16x16x4_f32(
                false, a, false, b, (short)0, dacc[nt], false, false);
        }
    }

    // bias + outer relu, store
    #pragma unroll
    for (int nt = 0; nt < 4; ++nt) {
        const float bias = b2[nt * 16 + l16];
        #pragma unroll
        for (int v = 0; v < 8; ++v) {
            float f = dacc[nt][v] + bias;
            const size_t r = (size_t)strip * 16 + v + half * 8;
            out[r * HIDDEN + nt * 16 + l16] = f > 0.f ? f : 0.f;
        }
    }
}

// mean-pool scatter: one wave per node, lane handles 2 channels
__global__ __launch_bounds__(256) void pool_scatter(
    const float* __restrict__ h, const int* __restrict__ batch,
    float* __restrict__ pool, float* __restrict__ cnt, int N)
{
    const int lane   = threadIdx.x & 31;
    const int wave   = (blockIdx.x * blockDim.x + threadIdx.x) >> 5;
    const int nWaves = (gridDim.x * blockDim.x) >> 5;
    const int c = lane * 2;
    for (int n = wave; n < N; n += nWaves) {
        const int g = batch[n];
        const float2 hv = *(const float2*)(h + (size_t)n * HIDDEN + c);
        float* p = pool + (size_t)g * HIDDEN + c;
        atomAddF(p,     hv.x);
        atomAddF(p + 1, hv.y);
        if (lane == 0) atomAddF(cnt + g, 1.0f);
    }
}

__global__ __launch_bounds__(256) void pool_finalize(
    const float* __restrict__ pool, const float* __restrict__ cnt,
    float* __restrict__ out)
{
    int i = blockIdx.x * blockDim.x + threadIdx.x;
    if (i < NGRAPHS * HIDDEN) {
        float c = cnt[i >> 6];
        c = c < 1.f ? 1.f : c;
        out[i] = pool[i] / c;
    }
}

extern "C" void kernel_launch(void* const* d_in, const int* in_sizes, int n_in,
                              void* d_out, int out_size, void* d_ws, size_t ws_size,
                              hipStream_t stream) {
    const float* x     = (const float*)d_in[0];
    const int*   ei    = (const int*)  d_in[1];   // [2,E] flat: src then dst
    const float* ea    = (const float*)d_in[2];
    const int*   batch = (const int*)  d_in[3];
    const float* ne_w  = (const float*)d_in[4];
    const float* ne_b  = (const float*)d_in[5];
    const float* ee_w  = (const float*)d_in[6];
    const float* ee_b  = (const float*)d_in[7];
    const float* c1_w1 = (const float*)d_in[8];
    const float* c1_b1 = (const float*)d_in[9];
    const float* c1_w2 = (const float*)d_in[10];
    const float* c1_b2 = (const float*)d_in[11];
    const float* c2_w1 = (const float*)d_in[12];
    const float* c2_b1 = (const float*)d_in[13];
    const float* c2_w2 = (const float*)d_in[14];
    const float* c2_b2 = (const float*)d_in[15];

    const int N = in_sizes[0];      // 100000 (multiple of 16)
    const int E = in_sizes[2];      // 1200000

    // workspace layout: h0 | agg | h1 | pool | cnt   (~77 MB)
    float* h0   = (float*)d_ws;
    float* agg  = h0  + (size_t)N * HIDDEN;
    float* h1   = agg + (size_t)N * HIDDEN;
    float* pool = h1  + (size_t)N * HIDDEN;
    float* cnt  = pool + (size_t)NGRAPHS * HIDDEN;

    hipMemsetAsync(agg,  0, (size_t)N * HIDDEN * sizeof(float), stream);
    hipMemsetAsync(pool, 0, ((size_t)NGRAPHS * HIDDEN + NGRAPHS) * sizeof(float), stream);

    const int nStrips  = (N + 15) / 16;
    const int mlpBlks  = (nStrips + 7) / 8;
    const int edgeBlks = (E + 7) / 8;

    node_encode<<<(N * HIDDEN + 255) / 256, 256, 0, stream>>>(x, ne_w, ne_b, h0, N);

    // layer 1
    edge_scatter<<<edgeBlks, 256, 0, stream>>>(h0, ei, ea, ee_w, ee_b, agg, E);
    gine_mlp_wmma<<<mlpBlks, 256, 0, stream>>>(h0, agg, c1_w1, c1_b1, c1_w2, c1_b2, h1, nStrips);

    // layer 2
    hipMemsetAsync(agg, 0, (size_t)N * HIDDEN * sizeof(float), stream);
    edge_scatter<<<edgeBlks, 256, 0, stream>>>(h1, ei, ea, ee_w, ee_b, agg, E);
    gine_mlp_wmma<<<mlpBlks, 256, 0, stream>>>(h1, agg, c2_w1, c2_b1, c2_w2, c2_b2, h0, nStrips);

    // global mean pool
    pool_scatter<<<(N + 7) / 8, 256, 0, stream>>>(h0, batch, pool, cnt, N);
    pool_finalize<<<(NGRAPHS * HIDDEN + 255) / 256, 256, 0, stream>>>(pool, cnt, (float*)d_out);
}